// InvGridSamplerDenominator_3066606649874
// MI455X (gfx1250) — compile-verified
//
#include <hip/hip_runtime.h>
#include <hip/hip_bf16.h>
#include <stdint.h>

// Problem shape (fixed by reference)
#define BSZ 8
#define NCH 32
#define H   512
#define W   512
#define HW        (H * W)          // 262144 = 1<<18
#define HW_SHIFT  18
#define HWV4      (HW / 4)         // 65536

typedef __attribute__((ext_vector_type(4))) float v4f;
typedef __attribute__((ext_vector_type(2))) float v2f;
typedef __attribute__((ext_vector_type(4))) int   v4i;

// Address-space-qualified pointer types for the gfx1250 async-LDS builtin.
typedef __attribute__((address_space(1))) v4i* gas_v4i;   // global
typedef __attribute__((address_space(3))) v4i* las_v4i;   // LDS

// ---- CDNA5 feature detection -------------------------------------------------
#if defined(__has_builtin)
#if __has_builtin(__builtin_amdgcn_global_load_async_to_lds_b128)
#define HAVE_ASYNC_LDS 1
#endif
#endif

static __device__ __forceinline__ void wait_async0() {
#if defined(__has_builtin) && __has_builtin(__builtin_amdgcn_s_wait_asynccnt)
  __builtin_amdgcn_s_wait_asynccnt(0);
#else
  asm volatile("s_wait_asynccnt 0" ::: "memory");
#endif
}

// ---- Kernel 1: zero the channel-0 plane of each batch (the accumulator) -----
__global__ void igsd_zero_ch0(float* __restrict__ out) {
  int tid = blockIdx.x * blockDim.x + threadIdx.x;        // over BSZ*HWV4
  int b = tid / HWV4;
  int r = tid - b * HWV4;
  v4f* p = (v4f*)(out + (size_t)b * NCH * HW) + r;
  v4f z = {0.f, 0.f, 0.f, 0.f};
  *p = z;
}

// ---- Kernel 2: bilinear scatter-add into channel-0 plane (f32 HW atomics) ---
__global__ void igsd_scatter(const float* __restrict__ inv_grid,
                             float* __restrict__ out) {
  int tid = blockIdx.x * blockDim.x + threadIdx.x;        // over BSZ*HW points
  int b = tid >> HW_SHIFT;

  // Prefetch the read stream a few cachelines ahead (gfx1250 global_prefetch_b8)
  __builtin_prefetch(inv_grid + 2u * (unsigned)tid + 512u, 0, 1);

  v2f g = ((const v2f*)inv_grid)[tid];                    // (gi_raw, gj_raw)

  // g = (inv_grid + 1) * 0.5 ; gi = clip(g*H + 1, 0, H+1-2eps)
  float gi = (g.x + 1.0f) * 0.5f * (float)H + 1.0f;
  float gj = (g.y + 1.0f) * 0.5f * (float)W + 1.0f;
  // f32(513 - 2e-10) == 513.0f, matching JAX float32 semantics
  gi = fminf(fmaxf(gi, 0.0f), (float)(H + 1));
  gj = fminf(fmaxf(gj, 0.0f), (float)(W + 1));

  int li = (int)gi;                                       // gi >= 0 -> trunc == floor
  int lj = (int)gj;

  float* plane = out + (size_t)b * NCH * HW;              // channel-0 accumulator

#pragma unroll
  for (int di = 0; di < 2; ++di) {
    int   ti = li + di;
    float wi = fmaxf(1.0f - fabsf(gi - (float)ti), 0.0f);
    bool  i_ok = (ti >= 1) & (ti <= H);
#pragma unroll
    for (int dj = 0; dj < 2; ++dj) {
      int   tj = lj + dj;
      float wj = fmaxf(1.0f - fabsf(gj - (float)tj), 0.0f);
      float wt = wi * wj;
      // Border cells (acc rows/cols outside [1,H]x[1,W]) are sliced away by the
      // reference -> skipping them is exact; also skip exact-zero weights.
      if (i_ok & (tj >= 1) & (tj <= W) & (wt != 0.0f)) {
        float* dst = plane + (size_t)(ti - 1) * W + (tj - 1);
        __hip_atomic_fetch_add(dst, wt, __ATOMIC_RELAXED, __HIP_MEMORY_SCOPE_AGENT);
      }
    }
  }
}

// ---- Kernel 3: broadcast channel 0 -> channels 1..31 (NT streaming stores) --
__global__ void igsd_broadcast(float* __restrict__ out) {
  int tid = blockIdx.x * blockDim.x + threadIdx.x;        // over BSZ*HWV4
  int b = tid / HWV4;
  int r = tid - b * HWV4;
  float* base = out + (size_t)b * NCH * HW;
  const v4f* src = (const v4f*)base + r;

  v4f v;
#if HAVE_ASYNC_LDS
  // Stage the channel-0 tile via the gfx1250 async DMA path (ASYNCcnt).
  __shared__ v4f stage[256];
  v4i* src_gen = (v4i*)const_cast<v4f*>(src);             // generic, const stripped
  v4i* dst_gen = (v4i*)&stage[threadIdx.x];               // generic LDS pointer
  __builtin_amdgcn_global_load_async_to_lds_b128(
      (gas_v4i)src_gen,                                   // addrspace(1) v4i*
      (las_v4i)dst_gen,                                   // addrspace(3) v4i*
      0, 0);
  wait_async0();
  v = stage[threadIdx.x];
#else
  v = *src;
#endif

#pragma unroll
  for (int c = 1; c < NCH; ++c) {
    v4f* dst = (v4f*)(base + (size_t)c * HW) + r;
    __builtin_nontemporal_store(v, dst);                  // TH=NT: keep L2 for the accumulator
  }
}

extern "C" void kernel_launch(void* const* d_in, const int* in_sizes, int n_in,
                              void* d_out, int out_size, void* d_ws, size_t ws_size,
                              hipStream_t stream) {
  (void)d_in; (void)in_sizes; (void)n_in; (void)d_ws; (void)ws_size; (void)out_size;
  const float* inv_grid = (const float*)d_in[1];          // inputs: [x, inv_grid]
  float* out = (float*)d_out;

  const int threads = 256;
  const int nZero  = BSZ * HWV4;                          // 524288 float4 stores
  const int nScat  = BSZ * HW;                            // 2097152 grid points
  const int nBcast = BSZ * HWV4;                          // 524288 float4 pixels

  igsd_zero_ch0 <<<nZero  / threads, threads, 0, stream>>>(out);
  igsd_scatter  <<<nScat  / threads, threads, 0, stream>>>(inv_grid, out);
  igsd_broadcast<<<nBcast / threads, threads, 0, stream>>>(out);
}